// DynamicDistributionShiftTracer_38268158608010
// MI455X (gfx1250) — compile-verified
//
#include <hip/hip_runtime.h>
#include <hip/hip_bf16.h>
#include <math.h>

// DynamicDistributionShiftTracer — fused single-pass kernel for gfx1250.
// B=32, C=64, T=2048, K=16, NSTAT=6. One workgroup per (b,c) row.
// Padded-LDS x row (no divergent loads); stats -> V_WMMA_F32_16X16X4_F32 x2
// (K padded 6->8, exact f32) -> softmax over taps (in-lane + one
// shfl_xor(16)) -> signed causal windowed sum (copysign-folded).

#define B_DIM  32
#define C_DIM  64
#define T_DIM  2048
#define K_TAPS 16
#define NST    6
#define BLOCK  256   // 8 waves (wave32)
#define PAD    16    // front zero pad: causal window reaches t-15, pool t-2

typedef __attribute__((ext_vector_type(2))) float v2f;
typedef __attribute__((ext_vector_type(8))) float v8f;

__global__ __launch_bounds__(BLOCK)
void ddst_kernel(const float* __restrict__ x,
                 const float* __restrict__ W,
                 const float* __restrict__ bias_g,
                 float* __restrict__ out)
{
    __shared__ float lxp[PAD + T_DIM + 4];   // zero-padded x row (~8.3 KB)
    __shared__ float redw[8];
    __shared__ float s_mean;

    const int tid  = threadIdx.x;
    const int bc   = blockIdx.x;          // b*C + c
    const int bIdx = bc / C_DIM;
    const int cIdx = bc % C_DIM;
    const int lane = tid & 31;
    const int wave = tid >> 5;
    const int n    = lane & 15;   // N (t within tile); also M index for A
    const int half = lane >> 4;

    // ---- stage x[b,c,:] into padded LDS (float4-coalesced), zero the pads ----
    const float4* xg  = reinterpret_cast<const float4*>(x + (size_t)bc * T_DIM);
    float4*       lx4 = reinterpret_cast<float4*>(lxp + PAD);  // 64B-aligned
    for (int i = tid; i < T_DIM / 4; i += BLOCK) lx4[i] = xg[i];
    if (tid < PAD) lxp[tid] = 0.f;
    if (tid < 4)   lxp[PAD + T_DIM + tid] = 0.f;
    __syncthreads();

    // ---- global mean over T: in-wave shuffle reduce + 8-partial combine ----
    float ps = 0.f;
    for (int i = tid; i < T_DIM; i += BLOCK) ps += lxp[PAD + i];
    #pragma unroll
    for (int o = 16; o > 0; o >>= 1) ps += __shfl_xor(ps, o, 32);
    if (lane == 0) redw[wave] = ps;
    __syncthreads();
    if (tid == 0) {
        float s = 0.f;
        #pragma unroll
        for (int w = 0; w < 8; ++w) s += redw[w];
        s_mean = s * (1.0f / T_DIM);
    }
    __syncthreads();
    const float gmean = s_mean;

    // ---- A-matrix = W_c [taps x stats], constant per block ----
    // ISA 32-bit A 16x4 layout: va.x = A[m][2*half], va.y = A[m][2*half+1]
    const float* Wc = W + (size_t)cIdx * K_TAPS * NST;   // [k][i]
    v2f a0, a1;
    a0.x = Wc[n * NST + 2 * half + 0];
    a0.y = Wc[n * NST + 2 * half + 1];
    a1.x = half ? 0.f : Wc[n * NST + 4];   // stats 4,5 in K'=0,1; K'=2,3 pad 0
    a1.y = half ? 0.f : Wc[n * NST + 5];

    float bias[8];
    #pragma unroll
    for (int j = 0; j < 8; ++j) bias[j] = bias_g[cIdx * K_TAPS + j + 8 * half];

    // ---- each wave owns 16 contiguous 16-t tiles ----
    const int tile0 = wave * 16;
    #pragma unroll 2
    for (int it = 0; it < 16; ++it) {
        const int t = (tile0 + it) * 16 + n;        // this lane's timestep
        const float* lc = lxp + PAD + t;

        // unconditional padded loads; pads are 0 (correct for avg pools)
        float vm2 = lc[-2], vm1 = lc[-1], v0 = lc[0], vp1 = lc[1], vp2 = lc[2];

        float s0 = vm2 + vm1 + v0 + vp1 + vp2;
        float s1 = vm2*vm2 + vm1*vm1 + v0*v0 + vp1*vp1 + vp2*vp2;
        float avg  = s0 * 0.2f;
        float avg2 = s1 * 0.2f;
        float stdv = sqrtf(fmaxf(avg2 - avg * avg, 1e-6f));
        float d3   = v0 - avg;  d3 = d3 * d3 * d3;

        // min/max must exclude padding: branch-free select center for OOB
        float m2 = (t >= 2)         ? vm2 : v0;
        float m1 = (t >= 1)         ? vm1 : v0;
        float p1 = (t <  T_DIM - 1) ? vp1 : v0;
        float p2 = (t <  T_DIM - 2) ? vp2 : v0;
        float mx = fmaxf(fmaxf(fmaxf(m2, m1), v0), fmaxf(p1, p2));
        float mn = fminf(fminf(fminf(m2, m1), v0), fminf(p1, p2));

        // B-matrix = stats [stat x t]; direct half-selects (no array indexing)
        v2f b0, b1;
        b0.x = half ? stdv : v0;      // K'=0 -> stat0(x)   | K'=2 -> stat2(std)
        b0.y = half ? d3   : gmean;   // K'=1 -> stat1(mean)| K'=3 -> stat3(3rd)
        b1.x = half ? 0.f  : mx;      // K'=0 -> stat4(max) | K'=2 -> pad 0
        b1.y = half ? 0.f  : mn;      // K'=1 -> stat5(min) | K'=3 -> pad 0

        // kernels[k][t] = W_c @ stats + bias (exact f32, two K=4 WMMAs)
        v8f acc;
        #pragma unroll
        for (int j = 0; j < 8; ++j) acc[j] = bias[j];
        acc = __builtin_amdgcn_wmma_f32_16x16x4_f32(
                  false, a0, false, b0, (short)0, acc, false, false);
        acc = __builtin_amdgcn_wmma_f32_16x16x4_f32(
                  false, a1, false, b1, (short)0, acc, false, false);
        // lane holds kernels[k = j + 8*half][t] for j = 0..7

        // softmax over k of |kernels|: 8 in-lane + one half-exchange
        float ab[8], mloc = -INFINITY;
        #pragma unroll
        for (int j = 0; j < 8; ++j) { ab[j] = fabsf(acc[j]); mloc = fmaxf(mloc, ab[j]); }
        mloc = fmaxf(mloc, __shfl_xor(mloc, 16, 32));
        float e[8], es = 0.f;
        #pragma unroll
        for (int j = 0; j < 8; ++j) { e[j] = __expf(ab[j] - mloc); es += e[j]; }
        es += __shfl_xor(es, 16, 32);
        float inv = 1.0f / es;

        // signed causal windowed sum: x[t + k - 15] (padded, unconditional).
        // e[j] >= 0, so e[j]*sign(kk) == copysign(e[j], kk), zeroed when kk==0
        const float* cw = lxp + PAD + t - (K_TAPS - 1) + 8 * half;
        float osum = 0.f;
        #pragma unroll
        for (int j = 0; j < 8; ++j) {
            float xv = cw[j];
            float kk = acc[j];
            float se = __builtin_copysignf(e[j], kk);   // v_bfi_b32
            se = (kk == 0.f) ? 0.f : se;                // sign(0) = 0
            osum += xv * se;                            // folds into v_fma
        }
        osum *= inv;
        osum += __shfl_xor(osum, 16, 32);

        if (half == 0)
            out[((size_t)bIdx * T_DIM + t) * C_DIM + cIdx] = osum;
    }
}

extern "C" void kernel_launch(void* const* d_in, const int* in_sizes, int n_in,
                              void* d_out, int out_size, void* d_ws, size_t ws_size,
                              hipStream_t stream) {
    const float* x = (const float*)d_in[0];   // [B,C,T]
    const float* W = (const float*)d_in[1];   // [C,K,6]
    const float* b = (const float*)d_in[2];   // [C,K]
    float* out = (float*)d_out;               // [B,T,C]
    (void)in_sizes; (void)n_in; (void)out_size; (void)d_ws; (void)ws_size;

    dim3 grid(B_DIM * C_DIM);   // 2048 workgroups, one per (b,c) row
    dim3 block(BLOCK);
    hipLaunchKernelGGL(ddst_kernel, grid, block, 0, stream, x, W, b, out);
}